// ComGNNBank_13365938225806
// MI455X (gfx1250) — compile-verified
//
#include <hip/hip_runtime.h>
#include <hip/hip_bf16.h>

#define N_NODES 50000
#define N_COMS  4
#define N_EDGES 800000
#define IN_DIM  128
#define COM_DIM 32

typedef __attribute__((ext_vector_type(2))) float v2f;
typedef __attribute__((ext_vector_type(8))) float v8f;

__device__ __forceinline__ float2 ld2(const float* p) { return *(const float2*)p; }

// Non-returning f32 atomic add -> global_atomic_add_f32 (relaxed, device scope).
__device__ __forceinline__ void atomAddF32(float* p, float v) {
  __hip_atomic_fetch_add(p, v, __ATOMIC_RELAXED, __HIP_MEMORY_SCOPE_AGENT);
}

// ---------------------------------------------------------------------------
// f32 GEMM tile kernel using V_WMMA_F32_16X16X4_F32 (exact fp32 path).
// One wave (32 lanes) computes a 16x16 tile of C = A @ B (+ bias).
// Fragment layouts (ISA 7.12.2):
//   A 16x4 (MxK): lane m = lane&15; lanes 0-15 hold K=0,1; lanes 16-31 K=2,3
//     -> per-lane contiguous K pair at k0 + 2*(lane>>4).
//   B 4x16 (KxN): mirrored; lane gives column N = lane&15, same K pair.
//   C/D 16x16:    lane gives column N = lane&15; VGPR r is row M = r + 8*(lane>>4).
// blockIdx.z selects a "community" slice via aCom/cCom element strides.
// ---------------------------------------------------------------------------
template <int K>
__global__ void __launch_bounds__(32)
gemm_wmma_kernel(const float* __restrict__ A, int lda, long aCom,
                 const float* __restrict__ B, int ldb,
                 const float* __restrict__ bias,
                 float* __restrict__ C, int ldc, long cCom) {
  const int lane = threadIdx.x;
  const int m = lane & 15, hs = lane >> 4;
  const int row0 = blockIdx.x * 16, col0 = blockIdx.y * 16;
  A += (long)blockIdx.z * aCom;
  C += (long)blockIdx.z * cCom;

  v8f acc;
  const float bv = bias ? bias[col0 + m] : 0.0f;
#pragma unroll
  for (int r = 0; r < 8; ++r) acc[r] = bv;

#pragma unroll
  for (int k0 = 0; k0 < K; k0 += 4) {
    const int kb = k0 + (hs << 1);            // this lane's contiguous K pair
    const float2 av = ld2(&A[(long)(row0 + m) * lda + kb]);
    v2f a, b;
    a[0] = av.x;
    a[1] = av.y;
    b[0] = B[(long)kb * ldb + col0 + m];
    b[1] = B[(long)(kb + 1) * ldb + col0 + m];
    acc = __builtin_amdgcn_wmma_f32_16x16x4_f32(false, a, false, b,
                                                (short)0, acc, false, false);
  }

#pragma unroll
  for (int r = 0; r < 8; ++r)
    C[(long)(row0 + r + (hs << 3)) * ldc + col0 + m] = acc[r];
}

// ---------------------------------------------------------------------------
// Fused second MLP GEMM: A_tile = relu(p + aggr + b1), then @ W2 + b2.
// Writes pre-BN z directly into its concatenated output slot [N, 128].
// ---------------------------------------------------------------------------
__global__ void __launch_bounds__(32)
mlp2_wmma_kernel(const float* __restrict__ p, long pCom,
                 const float* __restrict__ aggr,   // [NC][N][32]
                 const float* __restrict__ b1,
                 const float* __restrict__ W2,     // [32][32]
                 const float* __restrict__ b2,
                 float* __restrict__ outSlot) {    // [N][128]
  const int lane = threadIdx.x;
  const int m = lane & 15, hs = lane >> 4;
  const int mt = blockIdx.x, nt = blockIdx.y, k = blockIdx.z;
  const long node = (long)mt * 16 + m;
  const float* pk = p + (long)k * pCom;
  const float* ak = aggr + (long)k * N_NODES * COM_DIM;

  v8f acc;
  const float bv = b2[nt * 16 + m];
#pragma unroll
  for (int r = 0; r < 8; ++r) acc[r] = bv;

#pragma unroll
  for (int k0 = 0; k0 < COM_DIM; k0 += 4) {
    const int kb = k0 + (hs << 1);
    const float2 pv = ld2(&pk[node * COM_DIM + kb]);
    const float2 av = ld2(&ak[node * COM_DIM + kb]);
    float v0 = pv.x + av.x + b1[kb];
    float v1 = pv.y + av.y + b1[kb + 1];
    v0 = v0 > 0.0f ? v0 : 0.0f;
    v1 = v1 > 0.0f ? v1 : 0.0f;
    v2f a, b;
    a[0] = v0;
    a[1] = v1;
    b[0] = W2[kb * COM_DIM + nt * 16 + m];
    b[1] = W2[(kb + 1) * COM_DIM + nt * 16 + m];
    acc = __builtin_amdgcn_wmma_f32_16x16x4_f32(false, a, false, b,
                                                (short)0, acc, false, false);
  }

  const int col = k * COM_DIM + nt * 16 + m;
#pragma unroll
  for (int r = 0; r < 8; ++r)
    outSlot[(long)(mt * 16 + r + (hs << 3)) * IN_DIM + col] = acc[r];
}

// ---------------------------------------------------------------------------
// Edge aggregation in projected 32-dim space. One wave per edge; lane = feature.
// pCom==0: shared projection (layer 0). Otherwise per-community projections.
// ---------------------------------------------------------------------------
__global__ void __launch_bounds__(256)
edge_aggr_kernel(const int* __restrict__ ei,      // [2][E]
                 const float* __restrict__ ew,    // [NC][E]
                 const float* __restrict__ p, long pCom,
                 float* __restrict__ aggr) {      // [NC][N][32]
  const int lane = threadIdx.x & 31;
  const long e = (long)blockIdx.x * 8 + (threadIdx.x >> 5);
  if (e >= N_EDGES) return;
  const long s = ei[e];
  const long d = ei[N_EDGES + e];
  const long sOff = s * COM_DIM + lane;
  const long dOff = d * COM_DIM + lane;
  if (pCom == 0) {
    const float v = p[sOff];
#pragma unroll
    for (int k = 0; k < N_COMS; ++k) {
      const float w = ew[(long)k * N_EDGES + e];
      atomAddF32(&aggr[(long)k * N_NODES * COM_DIM + dOff], w * v);
    }
  } else {
#pragma unroll
    for (int k = 0; k < N_COMS; ++k) {
      const float v = p[(long)k * pCom + sOff];
      const float w = ew[(long)k * N_EDGES + e];
      atomAddF32(&aggr[(long)k * N_NODES * COM_DIM + dOff], w * v);
    }
  }
}

// ---------------------------------------------------------------------------
// Column-wise sum / sum-of-squares over a [N, 128] matrix (for BatchNorm).
// ---------------------------------------------------------------------------
__global__ void __launch_bounds__(128)
col_stats_kernel(const float* __restrict__ z, float* __restrict__ sums) {
  const int col = threadIdx.x;
  const int rowsPer = (N_NODES + gridDim.x - 1) / gridDim.x;
  long r0 = (long)blockIdx.x * rowsPer;
  long r1 = r0 + rowsPer;
  if (r1 > N_NODES) r1 = N_NODES;
  float s = 0.0f, ss = 0.0f;
  for (long r = r0; r < r1; ++r) {
    const float v = z[r * IN_DIM + col];
    s += v;
    ss += v * v;
  }
  atomAddF32(&sums[col], s);
  atomAddF32(&sums[128 + col], ss);
}

// ---------------------------------------------------------------------------
// In-place BatchNorm (batch statistics, population variance) + ReLU.
// gamma/beta are the shared [32] vectors, applied per community block.
// ---------------------------------------------------------------------------
__global__ void __launch_bounds__(256)
bn_relu_kernel(float* __restrict__ z,
               const float* __restrict__ sums,
               const float* __restrict__ gamma,
               const float* __restrict__ beta) {
  const long i = (long)blockIdx.x * 256 + threadIdx.x;
  if (i >= (long)N_NODES * IN_DIM) return;
  const int c = (int)(i & (IN_DIM - 1));
  const int f = c & (COM_DIM - 1);
  const float inv = 1.0f / (float)N_NODES;
  const float mu = sums[c] * inv;
  const float var = sums[128 + c] * inv - mu * mu;
  float v = (z[i] - mu) * rsqrtf(var + 1e-5f) * gamma[f] + beta[f];
  z[i] = v > 0.0f ? v : 0.0f;
}

// ---------------------------------------------------------------------------
extern "C" void kernel_launch(void* const* d_in, const int* in_sizes, int n_in,
                              void* d_out, int out_size, void* d_ws, size_t ws_size,
                              hipStream_t stream) {
  const float* x    = (const float*)d_in[0];
  const int*   ei   = (const int*)d_in[1];
  const float* ew   = (const float*)d_in[2];
  const float* l0W1 = (const float*)d_in[3];
  const float* l0b1 = (const float*)d_in[4];
  const float* l0W2 = (const float*)d_in[5];
  const float* l0b2 = (const float*)d_in[6];
  const float* l0g  = (const float*)d_in[7];
  const float* l0be = (const float*)d_in[8];
  const float* l1W1 = (const float*)d_in[9];
  const float* l1b1 = (const float*)d_in[10];
  const float* l1W2 = (const float*)d_in[11];
  const float* l1b2 = (const float*)d_in[12];
  const float* l1g  = (const float*)d_in[13];
  const float* l1be = (const float*)d_in[14];
  const float* encW = (const float*)d_in[15];
  const float* encb = (const float*)d_in[16];

  float* out0 = (float*)d_out;                       // x @ enc_W + enc_b
  float* out1 = out0 + (long)N_NODES * IN_DIM;       // layer-0 concat
  float* out2 = out1 + (long)N_NODES * IN_DIM;       // layer-1 concat

  const long comElems = (long)N_COMS * N_NODES * COM_DIM;   // 6.4M
  float* aggr = (float*)d_ws;          // [NC][N][32] aggregated projections
  float* p    = aggr + comElems;       // [NC][N][32] projections (L0: first N*32)
  float* sums = p + comElems;          // 256 floats: sum[128] | sumsq[128]

  const int MT = N_NODES / 16;         // 3125 row tiles (exact)
  const size_t aggrBytes = (size_t)comElems * sizeof(float);

  // ---- encoder: out0 = x @ enc_W + enc_b  (K=128) ----
  gemm_wmma_kernel<IN_DIM><<<dim3(MT, IN_DIM / 16, 1), 32, 0, stream>>>(
      x, IN_DIM, 0, encW, IN_DIM, encb, out0, IN_DIM, 0);

  // ---- layer 0 ----
  // p0 = x @ l0_W1 (shared across communities; linearity of GIN pre-activation)
  gemm_wmma_kernel<IN_DIM><<<dim3(MT, COM_DIM / 16, 1), 32, 0, stream>>>(
      x, IN_DIM, 0, l0W1, COM_DIM, nullptr, p, COM_DIM, 0);
  hipMemsetAsync(aggr, 0, aggrBytes, stream);
  hipMemsetAsync(sums, 0, 256 * sizeof(float), stream);
  edge_aggr_kernel<<<(N_EDGES + 7) / 8, 256, 0, stream>>>(ei, ew, p, 0, aggr);
  mlp2_wmma_kernel<<<dim3(MT, COM_DIM / 16, N_COMS), 32, 0, stream>>>(
      p, 0, aggr, l0b1, l0W2, l0b2, out1);
  col_stats_kernel<<<200, 128, 0, stream>>>(out1, sums);
  bn_relu_kernel<<<(N_NODES * IN_DIM + 255) / 256, 256, 0, stream>>>(
      out1, sums, l0g, l0be);

  // ---- layer 1 ----
  // p1_k = h0_k @ l1_W1   (h0_k = out1[:, k*32:(k+1)*32], K=32)
  gemm_wmma_kernel<COM_DIM><<<dim3(MT, COM_DIM / 16, N_COMS), 32, 0, stream>>>(
      out1, IN_DIM, COM_DIM, l1W1, COM_DIM, nullptr,
      p, COM_DIM, (long)N_NODES * COM_DIM);
  hipMemsetAsync(aggr, 0, aggrBytes, stream);
  hipMemsetAsync(sums, 0, 256 * sizeof(float), stream);
  edge_aggr_kernel<<<(N_EDGES + 7) / 8, 256, 0, stream>>>(
      ei, ew, p, (long)N_NODES * COM_DIM, aggr);
  mlp2_wmma_kernel<<<dim3(MT, COM_DIM / 16, N_COMS), 32, 0, stream>>>(
      p, (long)N_NODES * COM_DIM, aggr, l1b1, l1W2, l1b2, out2);
  col_stats_kernel<<<200, 128, 0, stream>>>(out2, sums);
  bn_relu_kernel<<<(N_NODES * IN_DIM + 255) / 256, 256, 0, stream>>>(
      out2, sums, l1g, l1be);

  (void)in_sizes; (void)n_in; (void)out_size; (void)ws_size;
}